// TimeAwareMultiHeadAttention_34780645163115
// MI455X (gfx1250) — compile-verified
//
#include <hip/hip_runtime.h>
#include <hip/hip_bf16.h>

typedef float v2f __attribute__((ext_vector_type(2)));
typedef float v8f __attribute__((ext_vector_type(8)));

#define Bsz 8
#define Lsz 256
#define Dsz 256
#define HN 8
#define HS 32
#define TK_STRIDE 36                    // 32 floats + 4 pad -> LDS bank-conflict avoidance
#define NEGV (-4294967295.0f)           // -2^32 + 1
#define INV_SQRT_HS 0.17677669529663687f

// ---------------------------------------------------------------------------
// Kernel 1: projections.  z=0: Q = queries*Qw^T + Qb
//                         z=1: K' = keys*Kw^T + Kb + abs_pos_K
//                         z=2: V' = keys*Vw^T + Vb + abs_pos_V
// One wave (32 threads) computes one 16x16 output tile via 64x WMMA f32 16x16x4.
// ---------------------------------------------------------------------------
__global__ __launch_bounds__(32) void proj_kernel(
    const float* __restrict__ queries, const float* __restrict__ keys,
    const float* __restrict__ Qw, const float* __restrict__ Qb,
    const float* __restrict__ Kw, const float* __restrict__ Kb,
    const float* __restrict__ Vw, const float* __restrict__ Vb,
    const float* __restrict__ posK, const float* __restrict__ posV,
    float* __restrict__ Qbuf, float* __restrict__ Kpbuf, float* __restrict__ Vpbuf)
{
    const int lane  = threadIdx.x;
    const int r0    = blockIdx.x * 16;     // rows over B*L = 2048
    const int c0    = blockIdx.y * 16;     // cols over D = 256
    const int which = blockIdx.z;          // uniform -> EXEC stays all-ones

    const float* X    = (which == 0) ? queries : keys;
    const float* W    = (which == 0) ? Qw : (which == 1) ? Kw : Vw;
    const float* bias = (which == 0) ? Qb : (which == 1) ? Kb : Vb;
    const float* pos  = (which == 1) ? posK : (which == 2) ? posV : nullptr;
    float* out        = (which == 0) ? Qbuf : (which == 1) ? Kpbuf : Vpbuf;

    const int mn = lane & 15;              // M index for A, N index for B
    const int kh = (lane >> 4) << 1;       // K half: 0 (lanes 0-15) or 2 (lanes 16-31)

    const float* xrow = X + (size_t)(r0 + mn) * Dsz;   // A[m][k]
    const float* wrow = W + (size_t)(c0 + mn) * Dsz;   // B[k][n] = W[n][k]

    v8f acc = {};
    for (int k0 = 0; k0 < Dsz; k0 += 4) {
        v2f a, bm;
        a.x  = xrow[k0 + kh];
        a.y  = xrow[k0 + kh + 1];
        bm.x = wrow[k0 + kh];
        bm.y = wrow[k0 + kh + 1];
        acc = __builtin_amdgcn_wmma_f32_16x16x4_f32(
            false, a, false, bm, (short)0, acc, false, false);
    }

    const int  colg = c0 + mn;
    const float bv  = bias[colg];
    const int  mAdd = (lane >> 4) << 3;    // upper half-lanes own M = r + 8
#pragma unroll
    for (int r = 0; r < 8; ++r) {
        const int row = r0 + r + mAdd;
        float v = acc[r] + bv;
        if (pos) v += pos[(size_t)row * Dsz + colg];
        out[(size_t)row * Dsz + colg] = v;
    }
}

// ---------------------------------------------------------------------------
// Kernel 2: per-head scores  S0[hb,i,j] = Q_[hb,i,:] . K'_[hb,j,:]   (K=32)
// One wave per 16x16 tile, 8x WMMA f32 16x16x4.
// ---------------------------------------------------------------------------
__global__ __launch_bounds__(32) void score_kernel(
    const float* __restrict__ Qbuf, const float* __restrict__ Kpbuf,
    float* __restrict__ S0)
{
    const int lane = threadIdx.x;
    const int i0 = blockIdx.x * 16;
    const int j0 = blockIdx.y * 16;
    const int hb = blockIdx.z;             // h*B + b  (head-major)
    const int h  = hb >> 3;                // / B (B==8)
    const int b  = hb & 7;

    const int mn = lane & 15;
    const int kh = (lane >> 4) << 1;
    const float* qrow = Qbuf  + ((size_t)(b * Lsz + i0 + mn) * Dsz) + h * HS;
    const float* krow = Kpbuf + ((size_t)(b * Lsz + j0 + mn) * Dsz) + h * HS;

    v8f acc = {};
#pragma unroll
    for (int k0 = 0; k0 < HS; k0 += 4) {
        v2f a, bm;
        a.x  = qrow[k0 + kh];
        a.y  = qrow[k0 + kh + 1];
        bm.x = krow[k0 + kh];
        bm.y = krow[k0 + kh + 1];
        acc = __builtin_amdgcn_wmma_f32_16x16x4_f32(
            false, a, false, bm, (short)0, acc, false, false);
    }

    const int mAdd = (lane >> 4) << 3;
    float* srow = S0 + (size_t)hb * Lsz * Lsz;
#pragma unroll
    for (int r = 0; r < 8; ++r) {
        const int i = i0 + r + mAdd;
        const int j = j0 + (lane & 15);
        srow[(size_t)i * Lsz + j] = acc[r];
    }
}

// ---------------------------------------------------------------------------
// Kernel 3: fused time-dot + mask + softmax + output.  One block per (hb, i).
// tK row tile (32KB) is staged into LDS with gfx1250 async-to-LDS copies
// (coalesced 128B-line transfers, ASYNCcnt-tracked, VGPR-bypassing); tV is
// streamed directly with coalesced loads.  Total HBM traffic ~1 GB (floor).
// ---------------------------------------------------------------------------
__global__ __launch_bounds__(256) void attn_kernel(
    const float* __restrict__ Qbuf, const float* __restrict__ Vpbuf,
    const float* __restrict__ S0,
    const float* __restrict__ tK, const float* __restrict__ tV,
    const unsigned char* __restrict__ time_mask,
    const unsigned char* __restrict__ attn_mask,
    float* __restrict__ outp)
{
    __shared__ float tkTile[Lsz * TK_STRIDE];   // 36,864 B, padded stride
    __shared__ float qi[HS];
    __shared__ float p[Lsz];
    __shared__ float red[Lsz];
    __shared__ float partial[8 * HS];

    const int t  = threadIdx.x;
    const int i  = blockIdx.x;
    const int hb = blockIdx.y;
    const int h  = hb >> 3;                // / B
    const int b  = hb & 7;
    const int tmb = hb >> 3;               // faithful repeat_interleave: hb / HN

    const size_t rowBase = (size_t)(b * Lsz + i) * Lsz * Dsz;   // [b,i,*,*]
    const int hoff = h * HS;

    // ---- async-stage tK[b,i, j=0..255, hoff..hoff+31] into LDS -----------
    // float4 chunk idx = it*256 + t ; j = idx>>3 ; c = idx&7.  Consecutive
    // lanes hit consecutive 16B chunks -> whole-cacheline coalesced copies.
    {
        const unsigned tkBase = (unsigned)(uintptr_t)&tkTile[0];
#pragma unroll
        for (int it = 0; it < 8; ++it) {
            const int idx = it * 256 + t;
            const int j = idx >> 3;
            const int c = (idx & 7) << 2;                       // float offset
            const unsigned long long g =
                (unsigned long long)(uintptr_t)(tK + rowBase + (size_t)j * Dsz + hoff + c);
            const unsigned l = tkBase + (unsigned)((j * TK_STRIDE + c) * 4);
            asm volatile("global_load_async_to_lds_b128 %0, %1, off"
                         :: "v"(l), "v"(g) : "memory");
        }
        asm volatile("s_wait_asynccnt 0x0" ::: "memory");
    }

    if (t < HS) qi[t] = Qbuf[(size_t)(b * Lsz + i) * Dsz + hoff + t];
    __syncthreads();

    // ---- phase 1: score for column j = t (tK dot from LDS) ---------------
    {
        const int j = t;
        const float* tkp = &tkTile[j * TK_STRIDE];
        float dot = 0.f;
#pragma unroll
        for (int k = 0; k < HS; k += 4) {
            const float4 v4 = *(const float4*)(tkp + k);
            dot += v4.x * qi[k] + v4.y * qi[k + 1] + v4.z * qi[k + 2] + v4.w * qi[k + 3];
        }
        float s = (S0[((size_t)hb * Lsz + i) * Lsz + j] + dot) * INV_SQRT_HS;
        const bool masked = (attn_mask[i * Lsz + j] != 0) ||
                            (time_mask[((size_t)tmb * Lsz + i) * Lsz + j] != 0);
        s = masked ? NEGV : s;
        p[t]   = s;
        red[t] = s;
    }
    __syncthreads();

    // ---- softmax over 256 columns ---------------------------------------
    for (int off = 128; off > 0; off >>= 1) {
        if (t < off) red[t] = fmaxf(red[t], red[t + off]);
        __syncthreads();
    }
    const float mx = red[0];
    __syncthreads();
    const float e = __expf(p[t] - mx);
    p[t]   = e;
    red[t] = e;
    __syncthreads();
    for (int off = 128; off > 0; off >>= 1) {
        if (t < off) red[t] += red[t + off];
        __syncthreads();
    }
    const float inv = 1.0f / red[0];
    __syncthreads();
    p[t] *= inv;
    __syncthreads();

    // ---- phase 2: out[s] = sum_j p[j] * (V'[j,s] + tV[i,j,s]) ------------
    const int w = t >> 5;                  // wave id 0..7
    const int s = t & 31;                  // lane = head-dim index (coalesced)
    float acc = 0.f;
    for (int j = w; j < Lsz; j += 8) {
        const float v = Vpbuf[(size_t)(b * Lsz + j) * Dsz + hoff + s]
                      + tV[rowBase + (size_t)j * Dsz + hoff + s];
        acc += p[j] * v;
    }
    partial[w * HS + s] = acc;
    __syncthreads();
    if (t < HS) {
        float o = 0.f;
#pragma unroll
        for (int ww = 0; ww < 8; ++ww) o += partial[ww * HS + t];
        outp[(size_t)(b * Lsz + i) * Dsz + hoff + t] = o;
    }
}

// ---------------------------------------------------------------------------
extern "C" void kernel_launch(void* const* d_in, const int* in_sizes, int n_in,
                              void* d_out, int out_size, void* d_ws, size_t ws_size,
                              hipStream_t stream) {
    const float*         queries   = (const float*)d_in[0];
    const float*         keys      = (const float*)d_in[1];
    const unsigned char* time_mask = (const unsigned char*)d_in[2];
    const unsigned char* attn_mask = (const unsigned char*)d_in[3];
    const float*         tK        = (const float*)d_in[4];
    const float*         tV        = (const float*)d_in[5];
    const float*         posK      = (const float*)d_in[6];
    const float*         posV      = (const float*)d_in[7];
    const float*         Qw        = (const float*)d_in[8];
    const float*         Qb        = (const float*)d_in[9];
    const float*         Kw        = (const float*)d_in[10];
    const float*         Kb        = (const float*)d_in[11];
    const float*         Vw        = (const float*)d_in[12];
    const float*         Vb        = (const float*)d_in[13];

    float* out = (float*)d_out;
    float* ws  = (float*)d_ws;

    const size_t NBLD = (size_t)Bsz * Lsz * Dsz;        // 524288
    float* Qbuf  = ws;
    float* Kpbuf = ws + NBLD;
    float* Vpbuf = ws + 2 * NBLD;
    float* S0    = ws + 3 * NBLD;                        // 64*256*256 floats

    // 1) projections (+ fold abs_pos into K', V')
    proj_kernel<<<dim3((Bsz * Lsz) / 16, Dsz / 16, 3), 32, 0, stream>>>(
        queries, keys, Qw, Qb, Kw, Kb, Vw, Vb, posK, posV, Qbuf, Kpbuf, Vpbuf);

    // 2) per-head Q.K'^T scores
    score_kernel<<<dim3(Lsz / 16, Lsz / 16, HN * Bsz), 32, 0, stream>>>(
        Qbuf, Kpbuf, S0);

    // 3) fused time-dot + softmax + output (bandwidth-bound part)
    attn_kernel<<<dim3(Lsz, HN * Bsz), 256, 0, stream>>>(
        Qbuf, Vpbuf, S0, tK, tV, time_mask, attn_mask, out);
}